// SimpleCGCNN_7292854469258
// MI455X (gfx1250) — compile-verified
//
#include <hip/hip_runtime.h>
#include <hip/hip_bf16.h>

// ---------------------------------------------------------------------------
// CGCNN forward for MI455X (gfx1250, wave32).
//
// Factorization: z @ W with z = [h_dst || h_src || e] splits into
//   PD[n] = [ h@Wf[0:64]   || h@Ws[0:64]   ]  (dst parts,  WMMA f32 16x16x4)
//   PS[n] = [ h@Wf[64:128] || h@Ws[64:128] ]  (src parts,  WMMA f32 16x16x4)
//   e @ W[128:160,:]  (edge-level, VALU with weights register-resident,
//                      edge_attr broadcast via v_readlane)
// Then per edge: m = sigmoid(.) * softplus(.), scatter-add into agg[dst].
// ---------------------------------------------------------------------------

#define ATOM_FEA 64
#define EDGE_DIM 32
#define N_CONV   3
#define BN_EPS   1e-5f

typedef __attribute__((ext_vector_type(2))) float v2f;
typedef __attribute__((ext_vector_type(8))) float v8f;

__device__ __forceinline__ v8f wmma_f32_4(v2f a, v2f b, v8f c) {
    // 8 args: (neg_a, A, neg_b, B, c_mod, C, reuse_a, reuse_b)
    return __builtin_amdgcn_wmma_f32_16x16x4_f32(false, a, false, b,
                                                 (short)0, c, false, false);
}

__device__ __forceinline__ float sigmoidf_(float x) {
    return 1.0f / (1.0f + __expf(-x));
}
__device__ __forceinline__ float softplusf_(float x) {
    // numerically stable logaddexp(x, 0)
    return fmaxf(x, 0.0f) + log1pf(__expf(-fabsf(x)));
}

// --------------------------------------------------------------------------
// h[n][f] = emb[x[n]][f]   (float4 vectorized: 16 threads per node row)
// --------------------------------------------------------------------------
__global__ void embed_kernel(const int* __restrict__ x,
                             const float* __restrict__ emb,
                             float* __restrict__ h, int n) {
    int t = blockIdx.x * blockDim.x + threadIdx.x;      // over n*16
    if (t < n * 16) {
        int node = t >> 4;
        int f4 = (t & 15) * 4;
        float4 v = *(const float4*)&emb[(size_t)x[node] * ATOM_FEA + f4];
        *(float4*)&h[(size_t)t * 4] = v;
    }
}

// --------------------------------------------------------------------------
// Node projections via WMMA f32 16x16x4.
// Block = 128 threads = 4 waves. Wave w computes:
//   w=0: h @ Wf[0:64]   -> PD cols  0..63     w=2: h @ Ws[0:64]   -> PD cols 64..127
//   w=1: h @ Wf[64:128] -> PS cols  0..63     w=3: h @ Ws[64:128] -> PS cols 64..127
// for 16 node rows (blockIdx.x * 16), all 64 output cols (4 N-tiles).
// A 16x4 f32 layout: lanes 0-15 M=lane {K=kk,kk+1}; lanes 16-31 {K=kk+2,kk+3}.
// B 4x16 f32 layout mirrors A's K assignment, N striped across lane group.
// C 16x16 f32: VGPR i -> M = i + 8*(lane>=16), N = lane&15.
// --------------------------------------------------------------------------
__global__ void __launch_bounds__(128)
node_proj_wmma(const float* __restrict__ h,
               const float* __restrict__ Wf,   // [160][64] this layer
               const float* __restrict__ Ws,   // [160][64] this layer
               float* __restrict__ PD,         // [n][128]
               float* __restrict__ PS,         // [n][128]
               int n) {
    const int lane = threadIdx.x & 31;
    const int w    = threadIdx.x >> 5;       // which of 4 projection matrices
    const int m0   = blockIdx.x * 16;        // node tile (n % 16 == 0 holds)
    const int lrow = lane & 15;
    const int khi  = (lane >> 4) * 2;        // K sub-offset for this half-wave

    const float* Wp = (w < 2 ? Wf : Ws) + (size_t)((w & 1) * 64) * ATOM_FEA;
    const float* hrow = h + (size_t)(m0 + lrow) * ATOM_FEA + khi;

    v8f acc[4] = {};

#pragma unroll
    for (int kk = 0; kk < ATOM_FEA; kk += 4) {
        v2f a;
        a.x = hrow[kk];
        a.y = hrow[kk + 1];
#pragma unroll
        for (int nt = 0; nt < 4; ++nt) {
            v2f b;
            b.x = Wp[(size_t)(kk + khi) * ATOM_FEA + nt * 16 + lrow];
            b.y = Wp[(size_t)(kk + khi + 1) * ATOM_FEA + nt * 16 + lrow];
            acc[nt] = wmma_f32_4(a, b, acc[nt]);
        }
    }

    float* Pout = (w & 1) ? PS : PD;
    const int coloff = (w >> 1) * 64;        // Wf -> 0..63, Ws -> 64..127
    const int rbase = m0 + (lane >> 4) * 8;
#pragma unroll
    for (int nt = 0; nt < 4; ++nt) {
#pragma unroll
        for (int i = 0; i < 8; ++i) {
            Pout[(size_t)(rbase + i) * 128 + coloff + nt * 16 + lrow] = acc[nt][i];
        }
    }
}

// --------------------------------------------------------------------------
// Edge kernel: one wave per edge (grid-stride). Lane covers features
// {lane, lane+32}. Edge-attr projection weights live in VGPRs (4 columns
// of 32 per lane); edge_attr broadcast with v_readlane (pure VALU loop,
// zero DS traffic).
// --------------------------------------------------------------------------
#define EBLK 256
__global__ void __launch_bounds__(EBLK)
edge_kernel(const int* __restrict__ src,
            const int* __restrict__ dst,
            const float* __restrict__ edge_attr,   // [E][32]
            const float* __restrict__ PD,          // [n][128]
            const float* __restrict__ PS,          // [n][128]
            const float* __restrict__ WfE,         // [32][64]
            const float* __restrict__ WsE,         // [32][64]
            const float* __restrict__ bf,          // [64]
            const float* __restrict__ bs,          // [64]
            float* __restrict__ agg,               // [n][64]
            int n_edges) {
    const int lane = threadIdx.x & 31;
    const int f0 = lane, f1 = lane + 32;

    // Preload this lane's 4 weight columns into registers (128 VGPRs).
    float wf0[EDGE_DIM], wf1[EDGE_DIM], ws0[EDGE_DIM], ws1[EDGE_DIM];
#pragma unroll
    for (int k = 0; k < EDGE_DIM; ++k) {
        wf0[k] = WfE[k * ATOM_FEA + f0];
        wf1[k] = WfE[k * ATOM_FEA + f1];
        ws0[k] = WsE[k * ATOM_FEA + f0];
        ws1[k] = WsE[k * ATOM_FEA + f1];
    }
    const float bf0 = bf[f0], bf1 = bf[f1];
    const float bs0 = bs[f0], bs1 = bs[f1];

    int wid = blockIdx.x * (EBLK / 32) + (threadIdx.x >> 5);
    int nwaves = gridDim.x * (EBLK / 32);

    for (int e = wid; e < n_edges; e += nwaves) {
        const int s = __builtin_nontemporal_load(&src[e]);
        const int d = __builtin_nontemporal_load(&dst[e]);
        const float ea =
            __builtin_nontemporal_load(&edge_attr[(size_t)e * EDGE_DIM + lane]);

        const float* pd = PD + (size_t)d * 128;   // 512B contiguous per edge
        const float* ps = PS + (size_t)s * 128;

        float af0 = pd[f0] + ps[f0] + bf0;
        float af1 = pd[f1] + ps[f1] + bf1;
        float as0 = pd[64 + f0] + ps[64 + f0] + bs0;
        float as1 = pd[64 + f1] + ps[64 + f1] + bs1;

#pragma unroll
        for (int k = 0; k < EDGE_DIM; ++k) {
            const float a = __int_as_float(
                __builtin_amdgcn_readlane(__float_as_int(ea), k));
            af0 = fmaf(a, wf0[k], af0);
            af1 = fmaf(a, wf1[k], af1);
            as0 = fmaf(a, ws0[k], as0);
            as1 = fmaf(a, ws1[k], as1);
        }

        const float m0 = sigmoidf_(af0) * softplusf_(as0);
        const float m1 = sigmoidf_(af1) * softplusf_(as1);
        atomicAdd(&agg[(size_t)d * ATOM_FEA + f0], m0);
        atomicAdd(&agg[(size_t)d * ATOM_FEA + f1], m1);
    }
}

// --------------------------------------------------------------------------
// h = relu( bn(agg) + h )   (eval-mode BN, float4 vectorized)
// --------------------------------------------------------------------------
__global__ void bn_res_relu(float* __restrict__ h,
                            const float* __restrict__ agg,
                            const float* __restrict__ gamma,
                            const float* __restrict__ beta,
                            const float* __restrict__ mean,
                            const float* __restrict__ var,
                            int n) {
    int t = blockIdx.x * blockDim.x + threadIdx.x;      // over n*16
    if (t < n * 16) {
        int f4 = (t & 15) * 4;
        float4 a = *(const float4*)&agg[(size_t)t * 4];
        float4 hv = *(const float4*)&h[(size_t)t * 4];
        float4 g = *(const float4*)&gamma[f4];
        float4 b = *(const float4*)&beta[f4];
        float4 mu = *(const float4*)&mean[f4];
        float4 vr = *(const float4*)&var[f4];
        float4 o;
        o.x = fmaxf((a.x - mu.x) * __frsqrt_rn(vr.x + BN_EPS) * g.x + b.x + hv.x, 0.0f);
        o.y = fmaxf((a.y - mu.y) * __frsqrt_rn(vr.y + BN_EPS) * g.y + b.y + hv.y, 0.0f);
        o.z = fmaxf((a.z - mu.z) * __frsqrt_rn(vr.z + BN_EPS) * g.z + b.z + hv.z, 0.0f);
        o.w = fmaxf((a.w - mu.w) * __frsqrt_rn(vr.w + BN_EPS) * g.w + b.w + hv.w, 0.0f);
        *(float4*)&h[(size_t)t * 4] = o;
    }
}

// --------------------------------------------------------------------------
// Global mean pool (sum + counts via atomics; divide in MLP kernel)
// --------------------------------------------------------------------------
__global__ void pool_kernel(const float* __restrict__ h,
                            const int* __restrict__ batch,
                            float* __restrict__ pooled,
                            float* __restrict__ counts,
                            int n) {
    int i = blockIdx.x * blockDim.x + threadIdx.x;      // over n*64
    if (i < n * ATOM_FEA) {
        int node = i >> 6;
        int f = i & 63;
        int g = batch[node];
        atomicAdd(&pooled[(size_t)g * ATOM_FEA + f], h[i]);
        if (f == 0) atomicAdd(&counts[g], 1.0f);
    }
}

// --------------------------------------------------------------------------
// Per-graph head: hid = relu(pooled/cnt @ W1 + b1); out = hid @ W2 + b2
// One block (128 threads) per graph.
// --------------------------------------------------------------------------
__global__ void __launch_bounds__(128)
mlp_kernel(const float* __restrict__ pooled,
           const float* __restrict__ counts,
           const float* __restrict__ W1,   // [64][128]
           const float* __restrict__ b1,   // [128]
           const float* __restrict__ W2,   // [128][1]
           const float* __restrict__ b2,   // [1]
           float* __restrict__ out) {
    __shared__ float sp[ATOM_FEA];
    __shared__ float sh[128];
    const int g = blockIdx.x;
    const int t = threadIdx.x;

    const float cnt = fmaxf(counts[g], 1.0f);
    if (t < ATOM_FEA) sp[t] = pooled[(size_t)g * ATOM_FEA + t] / cnt;
    __syncthreads();

    float acc = b1[t];
#pragma unroll 8
    for (int k = 0; k < ATOM_FEA; ++k)
        acc += sp[k] * W1[k * 128 + t];
    acc = fmaxf(acc, 0.0f);
    sh[t] = acc * W2[t];
    __syncthreads();

#pragma unroll
    for (int sft = 64; sft > 0; sft >>= 1) {
        if (t < sft) sh[t] += sh[t + sft];
        __syncthreads();
    }
    if (t == 0) out[g] = sh[0] + b2[0];
}

// --------------------------------------------------------------------------
extern "C" void kernel_launch(void* const* d_in, const int* in_sizes, int n_in,
                              void* d_out, int out_size, void* d_ws, size_t ws_size,
                              hipStream_t stream) {
    const int*   x         = (const int*)d_in[0];
    const int*   ei        = (const int*)d_in[1];     // [2][E]: row0=src, row1=dst
    const float* edge_attr = (const float*)d_in[2];
    const int*   batch     = (const int*)d_in[3];
    const float* emb       = (const float*)d_in[4];
    const float* Wf        = (const float*)d_in[5];   // [3][160][64]
    const float* bf        = (const float*)d_in[6];   // [3][64]
    const float* Ws        = (const float*)d_in[7];
    const float* bs        = (const float*)d_in[8];
    const float* bn_gamma  = (const float*)d_in[9];
    const float* bn_beta   = (const float*)d_in[10];
    const float* bn_mean   = (const float*)d_in[11];
    const float* bn_var    = (const float*)d_in[12];
    const float* W1        = (const float*)d_in[13];
    const float* b1        = (const float*)d_in[14];
    const float* W2        = (const float*)d_in[15];
    const float* b2        = (const float*)d_in[16];
    float* out = (float*)d_out;

    const int N = in_sizes[0];              // 50000
    const int E = in_sizes[2] / EDGE_DIM;   // 1600000
    const int G = out_size;                 // 512

    const int* src = ei;
    const int* dst = ei + E;

    // workspace carve-up (floats)
    float* ws     = (float*)d_ws;
    float* h      = ws;                                    // N*64
    float* PD     = h + (size_t)N * ATOM_FEA;              // N*128
    float* PS     = PD + (size_t)N * 128;                  // N*128
    float* agg    = PS + (size_t)N * 128;                  // N*64
    float* pooled = agg + (size_t)N * ATOM_FEA;            // G*64
    float* counts = pooled + (size_t)G * ATOM_FEA;         // G

    const int vec_blocks = (N * 16 + 255) / 256;

    embed_kernel<<<vec_blocks, 256, 0, stream>>>(x, emb, h, N);

    for (int l = 0; l < N_CONV; ++l) {
        const float* Wfl = Wf + (size_t)l * 160 * ATOM_FEA;
        const float* Wsl = Ws + (size_t)l * 160 * ATOM_FEA;

        node_proj_wmma<<<N / 16, 128, 0, stream>>>(h, Wfl, Wsl, PD, PS, N);

        hipMemsetAsync(agg, 0, (size_t)N * ATOM_FEA * sizeof(float), stream);

        edge_kernel<<<4096, EBLK, 0, stream>>>(
            src, dst, edge_attr, PD, PS,
            Wfl + 128 * ATOM_FEA, Wsl + 128 * ATOM_FEA,
            bf + l * ATOM_FEA, bs + l * ATOM_FEA,
            agg, E);

        bn_res_relu<<<vec_blocks, 256, 0, stream>>>(
            h, agg,
            bn_gamma + l * ATOM_FEA, bn_beta + l * ATOM_FEA,
            bn_mean + l * ATOM_FEA, bn_var + l * ATOM_FEA, N);
    }

    hipMemsetAsync(pooled, 0, ((size_t)G * ATOM_FEA + G) * sizeof(float), stream);
    pool_kernel<<<(N * ATOM_FEA + 255) / 256, 256, 0, stream>>>(h, batch, pooled, counts, N);
    mlp_kernel<<<G, 128, 0, stream>>>(pooled, counts, W1, b1, W2, b2, out);
}